// CollageRenderer_30528627540019
// MI455X (gfx1250) — compile-verified
//
#include <hip/hip_runtime.h>
#include <hip/hip_bf16.h>
#include <math.h>

#define NSHAPES 256
#define CANVAS_S 512
#define TEXS 256
#define TEXPIX (TEXS * TEXS)

typedef __attribute__((ext_vector_type(4))) unsigned int uint32x4_t_;
typedef __attribute__((ext_vector_type(8))) int int32x8_t_;
typedef __attribute__((ext_vector_type(4))) int int32x4_t_;

// ---------------------------------------------------------------------------
// Kernel 1: per-shape parameter precompute (256 shapes, run once per launch).
// Layout per shape (16 floats = 64B):
//   [0]=x0 [1]=y0 [2]=w-1 [3]=h-1
//   [4]=cos [5]=sin [6]=cx [7]=cy
//   [8]=TEX/w [9]=TEX/h [10]=w-0.5 [11]=h-0.5
//   [12..14]=color RGB [15]=pad
// ---------------------------------------------------------------------------
__global__ void prep_params_kernel(const float* __restrict__ xr,
                                   const float* __restrict__ yr,
                                   const float* __restrict__ sxr,
                                   const float* __restrict__ syr,
                                   const float* __restrict__ ang,
                                   const float* __restrict__ col,
                                   float* __restrict__ params) {
  int i = blockIdx.x * blockDim.x + threadIdx.x;
  if (i >= NSHAPES) return;
  const float S = (float)CANVAS_S;
  float tx  = (1.0f / (1.0f + expf(-xr[i])))  * (S - 1.0f);
  float ty  = (1.0f / (1.0f + expf(-yr[i])))  * (S - 1.0f);
  float scx = (1.0f / (1.0f + expf(-sxr[i]))) * 0.6f + 0.2f;
  float scy = (1.0f / (1.0f + expf(-syr[i]))) * 0.6f + 0.2f;
  float w = rintf(fmaxf(S * scx, 1.0f));   // rintf = round-half-even, matches jnp.round
  float h = rintf(fmaxf(S * scy, 1.0f));
  float x0 = rintf(tx - w * 0.5f);
  float y0 = rintf(ty - h * 0.5f);
  float th = ang[i] * 0.017453292519943295f;
  float c = cosf(th), s = sinf(th);
  float* p = params + i * 16;
  p[0] = x0;               p[1] = y0;
  p[2] = w - 1.0f;         p[3] = h - 1.0f;
  p[4] = c;                p[5] = s;
  p[6] = (w - 1.0f) * 0.5f; p[7] = (h - 1.0f) * 0.5f;
  p[8] = (float)TEXS / w;  p[9] = (float)TEXS / h;
  p[10] = w - 0.5f;        p[11] = h - 0.5f;
  p[12] = col[i * 3 + 0];  p[13] = col[i * 3 + 1];
  p[14] = col[i * 3 + 2];  p[15] = 0.0f;
}

// ---------------------------------------------------------------------------
// Kernel 2: planar (4,256,256) -> interleaved RGBA float4 so each bilinear
// corner is one global_load_b128 instead of four b32 gathers.
// ---------------------------------------------------------------------------
__global__ void interleave_tex_kernel(const float* __restrict__ tex,
                                      float4* __restrict__ rgba) {
  int idx = blockIdx.x * blockDim.x + threadIdx.x;
  if (idx >= TEXPIX) return;
  rgba[idx] = make_float4(tex[idx],
                          tex[idx + TEXPIX],
                          tex[idx + 2 * TEXPIX],
                          tex[idx + 3 * TEXPIX]);
}

__device__ __forceinline__ float4 lerp4(float4 a, float4 b, float t) {
  return make_float4(a.x + (b.x - a.x) * t,
                     a.y + (b.y - a.y) * t,
                     a.z + (b.z - a.z) * t,
                     a.w + (b.w - a.w) * t);
}

// ---------------------------------------------------------------------------
// Kernel 3: per-pixel composite. Shape params staged to LDS once per block
// via the Tensor Data Mover; then an order-preserving block-level shape cull
// (wave32 ballot + cross-wave scan) so each 32x8 tile only walks the shapes
// whose bbox intersects it (~75 of 256 on average).
// ---------------------------------------------------------------------------
template <int INTERLEAVED>
__global__ __launch_bounds__(256) void render_kernel(
    const float* __restrict__ params_g,
    const float* __restrict__ texp,
    const float4* __restrict__ texi,
    float* __restrict__ out) {
  __shared__ float sp[NSHAPES * 16];        // 16 KB param table
  __shared__ unsigned short slist[NSHAPES]; // culled, order-preserving shape list
  __shared__ int wtot[8];                   // per-wave hit counts
  int tid = threadIdx.y * 32 + threadIdx.x;

#if __has_builtin(__builtin_amdgcn_tensor_load_to_lds) && __has_builtin(__builtin_amdgcn_s_wait_tensorcnt)
  if (tid < 32) {  // wave 0 issues the TDM DMA (wave-granular branch)
    unsigned long long ga = (unsigned long long)(uintptr_t)params_g;
    unsigned int lds_base = (unsigned int)(uintptr_t)(&sp[0]);  // low 32 bits of generic = LDS offset
    // D# group 0: count=1, lds_addr, global_addr[56:0], type=2
    uint32x4_t_ g0;
    g0.x = 1u;
    g0.y = lds_base;
    g0.z = (unsigned int)(ga & 0xFFFFFFFFu);
    g0.w = (unsigned int)((ga >> 32) & 0x01FFFFFFu) | (2u << 30);
    // D# group 1: mask=0, data_size=4B(enc 2), tensor_dim0=4096, tensor_dim1=1,
    //             tile_dim0=4096, tile_dim1=1, strides=4096  (one 16KB row)
    int32x8_t_ g1 = {0, 0, 0, 0, 0, 0, 0, 0};
    g1[0] = 0x00020000;           // [17:16] data_size = 2 -> 4 bytes
    g1[1] = (0x1000 << 16);       // tensor_dim0[15:0] = 4096 in bits[63:48]
    g1[2] = (1 << 16);            // tensor_dim1[15:0] = 1 (word2[31:16])
    g1[3] = (0x1000 << 16);       // tile_dim0 = 4096 in bits[127:112]
    g1[4] = 1;                    // tile_dim1 = 1, tile_dim2 = 0
    g1[5] = 0x1000;               // tensor_dim0_stride = 4096
    g1[6] = (0x1000 << 16);       // tensor_dim1_stride lo16 = 4096
    int32x4_t_ gz4 = {0, 0, 0, 0};
    int32x8_t_ gz8 = {0, 0, 0, 0, 0, 0, 0, 0};
    __builtin_amdgcn_tensor_load_to_lds(g0, g1, gz4, gz4, gz8, 0);
    __builtin_amdgcn_s_wait_tensorcnt(0);
  }
  __syncthreads();
#else
  for (int j = tid; j < NSHAPES * 16; j += 256) sp[j] = params_g[j];
  __syncthreads();
#endif

  // ---- order-preserving block-level cull: thread tid tests shape tid ----
  int bx0 = blockIdx.x * 32;
  int by0 = blockIdx.y * 8;
  {
    const float4* pb = (const float4*)(&sp[tid * 16]);
    float4 b0 = pb[0];  // x0, y0, w-1, h-1
    bool hit = ((float)bx0        <= b0.x + b0.z) &&
               ((float)(bx0 + 31) >= b0.x)        &&
               ((float)by0        <= b0.y + b0.w) &&
               ((float)(by0 + 7)  >= b0.y);
    int lane = tid & 31;
    int wv   = tid >> 5;
    unsigned mm = (unsigned)__ballot((int)hit);          // wave32: low 32 bits
    int prefix = __popc(mm & ((1u << lane) - 1u));       // lane=0 -> mask 0
    if (lane == 0) wtot[wv] = __popc(mm);
    __syncthreads();
    int base = 0;
    for (int k = 0; k < wv; ++k) base += wtot[k];
    if (hit) slist[base + prefix] = (unsigned short)tid;
  }
  int total = 0;
  for (int k = 0; k < 8; ++k) total += wtot[k];
  __syncthreads();

  int x = bx0 + threadIdx.x;
  int y = by0 + threadIdx.y;
  float px = (float)x, py = (float)y;
  float cr = 1.0f, cg = 1.0f, cb = 1.0f;  // canvas init = ones

  for (int j = 0; j < total; ++j) {
    int i = (int)slist[j];                   // wave-uniform -> LDS broadcast
    const float4* p4 = (const float4*)(&sp[i * 16]);
    float4 a0 = p4[0];                       // x0, y0, w-1, h-1
    float u = px - a0.x;
    float v = py - a0.y;
    if (u < 0.0f || v < 0.0f || u > a0.z || v > a0.w) continue;  // paste_ok
    float4 a1 = p4[1];                       // cos, sin, cx, cy
    float4 a2 = p4[2];                       // TEX/w, TEX/h, w-0.5, h-0.5
    float du = u - a1.z, dv = v - a1.w;
    float su =  a1.x * du + a1.y * dv + a1.z;
    float sv = -a1.y * du + a1.x * dv + a1.w;
    if (!(su > -0.5f && su < a2.z && sv > -0.5f && sv < a2.w)) continue;  // rot_ok
    float tsx = (su + 0.5f) * a2.x - 0.5f;
    float tsy = (sv + 0.5f) * a2.y - 0.5f;
    float xf = floorf(tsx), yf = floorf(tsy);
    float wx = tsx - xf, wy = tsy - yf;
    int xi0 = (int)xf, yi0 = (int)yf;
    int xi1 = xi0 + 1, yi1 = yi0 + 1;
    xi0 = xi0 < 0 ? 0 : (xi0 > TEXS - 1 ? TEXS - 1 : xi0);
    xi1 = xi1 < 0 ? 0 : (xi1 > TEXS - 1 ? TEXS - 1 : xi1);
    yi0 = yi0 < 0 ? 0 : (yi0 > TEXS - 1 ? TEXS - 1 : yi0);
    yi1 = yi1 < 0 ? 0 : (yi1 > TEXS - 1 ? TEXS - 1 : yi1);

    float4 v00, v01, v10, v11;
    if (INTERLEAVED) {
      v00 = texi[yi0 * TEXS + xi0];
      v01 = texi[yi0 * TEXS + xi1];
      v10 = texi[yi1 * TEXS + xi0];
      v11 = texi[yi1 * TEXS + xi1];
    } else {
      int b00 = yi0 * TEXS + xi0, b01 = yi0 * TEXS + xi1;
      int b10 = yi1 * TEXS + xi0, b11 = yi1 * TEXS + xi1;
      v00 = make_float4(texp[b00], texp[b00 + TEXPIX], texp[b00 + 2 * TEXPIX], texp[b00 + 3 * TEXPIX]);
      v01 = make_float4(texp[b01], texp[b01 + TEXPIX], texp[b01 + 2 * TEXPIX], texp[b01 + 3 * TEXPIX]);
      v10 = make_float4(texp[b10], texp[b10 + TEXPIX], texp[b10 + 2 * TEXPIX], texp[b10 + 3 * TEXPIX]);
      v11 = make_float4(texp[b11], texp[b11 + TEXPIX], texp[b11 + 2 * TEXPIX], texp[b11 + 3 * TEXPIX]);
    }
    float4 top = lerp4(v00, v01, wx);
    float4 bot = lerp4(v10, v11, wx);
    float4 smp = lerp4(top, bot, wy);

    float4 a3 = p4[3];                       // color RGB
    float alpha = smp.w;
    float om = 1.0f - alpha;
    cr = cr * om + smp.x * a3.x * alpha;
    cg = cg * om + smp.y * a3.y * alpha;
    cb = cb * om + smp.z * a3.z * alpha;
  }

  int o = y * CANVAS_S + x;
  out[o]                           = fminf(fmaxf(cr, 0.0f), 1.0f);
  out[o + CANVAS_S * CANVAS_S]     = fminf(fmaxf(cg, 0.0f), 1.0f);
  out[o + 2 * CANVAS_S * CANVAS_S] = fminf(fmaxf(cb, 0.0f), 1.0f);
}

// ---------------------------------------------------------------------------
extern "C" void kernel_launch(void* const* d_in, const int* in_sizes, int n_in,
                              void* d_out, int out_size, void* d_ws, size_t ws_size,
                              hipStream_t stream) {
  const float* xr  = (const float*)d_in[0];
  const float* yr  = (const float*)d_in[1];
  const float* sxr = (const float*)d_in[2];
  const float* syr = (const float*)d_in[3];
  const float* ang = (const float*)d_in[4];
  const float* col = (const float*)d_in[5];
  const float* tex = (const float*)d_in[6];
  float* out = (float*)d_out;

  const size_t PARAM_BYTES = (size_t)NSHAPES * 16 * sizeof(float);   // 16 KB
  const size_t TEXI_BYTES  = (size_t)TEXPIX * sizeof(float4);        // 1 MB
  float*  params = (float*)d_ws;
  float4* texi   = (float4*)((char*)d_ws + PARAM_BYTES);
  bool interleaved = ws_size >= PARAM_BYTES + TEXI_BYTES;

  prep_params_kernel<<<1, 256, 0, stream>>>(xr, yr, sxr, syr, ang, col, params);

  dim3 grid(CANVAS_S / 32, CANVAS_S / 8);
  dim3 block(32, 8);
  if (interleaved) {
    interleave_tex_kernel<<<TEXPIX / 256, 256, 0, stream>>>(tex, texi);
    render_kernel<1><<<grid, block, 0, stream>>>(params, tex, texi, out);
  } else {
    render_kernel<0><<<grid, block, 0, stream>>>(params, tex, texi, out);
  }
}